// RippleNet4SR_85727547228323
// MI455X (gfx1250) — compile-verified
//
#include <hip/hip_runtime.h>
#include <hip/hip_bf16.h>
#include <math.h>

// Model dims (fixed by reference)
#define Bn 8
#define Ln 50
#define Mn 32
#define Dn 64
#define HOPn 2

typedef float v8f __attribute__((ext_vector_type(8)));
typedef float v2f __attribute__((ext_vector_type(2)));
typedef unsigned int u32x4 __attribute__((ext_vector_type(4)));
typedef int i32x4 __attribute__((ext_vector_type(4)));
typedef int i32x8 __attribute__((ext_vector_type(8)));

// ---------------- WMMA helpers (V_WMMA_F32_16X16X4_F32) -----------------
// A 16x4 layout: lanes 0-15 hold M=0..15 with VGPR{0,1}=K{0,1}; lanes 16-31
// hold K{2,3}.  B mirrors A with N in place of M.  C/D: lane = n + 16*(m>=8),
// vgpr v -> m = v + 8*(lane>=16).

__device__ __forceinline__ v8f wmma4(v2f a, v2f b, v8f c) {
  return __builtin_amdgcn_wmma_f32_16x16x4_f32(false, a, false, b, (short)0, c,
                                               false, false);
}

// acc += X[rbase..rbase+15][0..K) @ W^T   with B(k,n) = W[(cbase+n)*sw + k]
__device__ __forceinline__ v8f wmma_xwT(const float* X, int sx, int rbase,
                                        const float* W, int sw, int cbase,
                                        int K, v8f acc) {
  const int lane = threadIdx.x & 31;
  const int hi = (lane >> 4) & 1;
  const int mn = lane & 15;
  const float* xr = X + (size_t)(rbase + mn) * sx;
  const float* wr = W + (size_t)(cbase + mn) * sw;
  for (int k0 = 0; k0 < K; k0 += 4) {
    int ka = k0 + 2 * hi;
    v2f a, b;
    a.x = xr[ka]; a.y = xr[ka + 1];
    b.x = wr[ka]; b.y = wr[ka + 1];
    acc = wmma4(a, b, acc);
  }
  return acc;
}

// acc += X[rbase..][0..K) @ V            with B(k,n) = V[k*sv + cbase+n]
__device__ __forceinline__ v8f wmma_xw(const float* X, int sx, int rbase,
                                       const float* V, int sv, int cbase,
                                       int K, v8f acc) {
  const int lane = threadIdx.x & 31;
  const int hi = (lane >> 4) & 1;
  const int mn = lane & 15;
  const float* xr = X + (size_t)(rbase + mn) * sx;
  for (int k0 = 0; k0 < K; k0 += 4) {
    int ka = k0 + 2 * hi;
    v2f a, b;
    a.x = xr[ka]; a.y = xr[ka + 1];
    b.x = V[(size_t)ka * sv + cbase + mn];
    b.y = V[(size_t)(ka + 1) * sv + cbase + mn];
    acc = wmma4(a, b, acc);
  }
  return acc;
}

__device__ __forceinline__ void wmma_store(float* O, int so, int rbase,
                                           int cbase, v8f acc, int rmax) {
  const int lane = threadIdx.x & 31;
  const int hi = (lane >> 4) & 1;
  const int n = lane & 15;
#pragma unroll
  for (int v = 0; v < 8; ++v) {
    int r = rbase + v + 8 * hi;
    if (r < rmax) O[(size_t)r * so + cbase + n] = acc[v];
  }
}

__device__ __forceinline__ float wred_sum(float v) {
  for (int off = 16; off; off >>= 1) v += __shfl_xor(v, off, 32);
  return v;
}
__device__ __forceinline__ float wred_max(float v) {
  for (int off = 16; off; off >>= 1) v = fmaxf(v, __shfl_xor(v, off, 32));
  return v;
}
__device__ __forceinline__ float lrelu(float x) { return x > 0.f ? x : 0.01f * x; }

// ---------------- TDM: DMA one 64x64 f32 tile global -> LDS -------------
// D# per CDNA5 ISA ch.8: group0 = {count=1 | lds_addr | global_addr | type=2}
// group1 = {data_size=4B, pad_enable, pad_interval=64 DW (code 5),
//           pad_amount=2 DW (code 1), tensor 64x64, tile 64x64, stride0=64}
// The LDS pad gives row stride 66 floats (bank-conflict-free, 8B aligned).
// This toolchain's builtin is the 6-arg clang-23 form:
//   (u32x4 g0, i32x8 g1, i32x4 g2, i32x4 g3, i32x8 g4, i32 cpol)
#define RSTRIDE 66

__device__ __forceinline__ void tdm_load_mat64(const float* gsrc,
                                               unsigned lds_byte_off) {
  unsigned long long ga = (unsigned long long)(uintptr_t)gsrc;
  u32x4 g0;
  g0[0] = 1u;                                   // count=1 (valid, user mode)
  g0[1] = lds_byte_off;                         // lds_addr (bytes)
  g0[2] = (unsigned)ga;                         // global_addr[31:0]
  g0[3] = ((unsigned)(ga >> 32) & 0x01FFFFFFu)  // global_addr[56:32]
        | (2u << 30);                           // type = 2 ("image")
  i32x8 g1;
  g1[0] = (2 << 16)                             // data_size: 4 bytes
        | (1 << 20)                             // pad_enable
        | (5 << 22)                             // pad_interval: 64 DWORDs
        | (1 << 25);                            // pad_amount: 2 DWORDs
  g1[1] = 64 << 16;                             // tensor_dim0 = 64 (bits 79:48)
  g1[2] = 64 << 16;                             // tensor_dim1 = 64 (bits 111:80)
  g1[3] = 64 << 16;                             // tile_dim0 = 64 (bits 127:112)
  g1[4] = 64;                                   // tile_dim1 = 64, tile_dim2 = 0
  g1[5] = 64;                                   // tensor_dim0_stride = 64
  g1[6] = 0;                                    // stride0 hi / stride1 lo
  g1[7] = 0;                                    // stride1 hi
  i32x4 gz4 = {0, 0, 0, 0};                     // 2D tile: dims 2..4 unused
  i32x8 gz8 = {0, 0, 0, 0, 0, 0, 0, 0};
  __builtin_amdgcn_tensor_load_to_lds(g0, g1, gz4, gz4, gz8, 0);
}

// ---------------- Kernel 1: ripple hops (TDM-staged streaming) ----------
// One wave per (b,l).  Triple-buffered LDS tiles; up to 3 TDM transfers in
// flight (TENSORcnt) while the wave computes the 64x64 matvec out of LDS.
__global__ __launch_bounds__(32) void k_ripple(const float* __restrict__ ent,
                                               const float* __restrict__ rel,
                                               const float* __restrict__ tw,
                                               const int* __restrict__ seq,
                                               const int* __restrict__ mh,
                                               const int* __restrict__ mr,
                                               const int* __restrict__ mt,
                                               float* __restrict__ us) {
  const int bl = blockIdx.x;            // b*L + l
  const int lane = threadIdx.x;         // single wave32
  __shared__ float rmat[3][RSTRIDE * 64];
  __shared__ float hbuf[64], tmp[64];

  const int base0 = bl * HOPn * Mn;     // memory ids, flattened over both hops

  // item0 / item / o_sum live in registers: lane owns elements {lane, lane+32}
  float i0a = ent[(size_t)seq[bl] * Dn + lane];
  float i0b = ent[(size_t)seq[bl] * Dn + lane + 32];
  float ia = i0a, ib = i0b;
  float osa = 0.f, osb = 0.f;

  // prologue: start DMA for tiles g = 0,1,2
  for (int g = 0; g < 3; ++g)
    tdm_load_mat64(rel + (size_t)mr[base0 + g] * (Dn * Dn),
                   (unsigned)(uintptr_t)&rmat[g][0]);

  for (int hop = 0; hop < HOPn; ++hop) {
    float myscore = 0.f;                // lane m keeps score[m]
    float oa = 0.f, ob = 0.f;
    for (int m = 0; m < Mn; ++m) {
      const int g = hop * Mn + m;
      // stage head vector for this memory (coalesced, 2 loads per lane)
      const float* hp = ent + (size_t)mh[base0 + g] * Dn;
      hbuf[lane] = hp[lane];
      hbuf[lane + 32] = hp[lane + 32];
      // wait until tile g is resident (keep 2 younger DMAs in flight)
      if (g >= 61) __builtin_amdgcn_s_wait_tensorcnt(0);
      else         __builtin_amdgcn_s_wait_tensorcnt(2);
      asm volatile("" ::: "memory");
      __syncthreads();                  // wave-local: orders hbuf stores too
      // Rh rows {lane, lane+32} from LDS tile (stride 66, conflict-free)
      const float* r0 = &rmat[g % 3][lane * RSTRIDE];
      const float* r1 = &rmat[g % 3][(lane + 32) * RSTRIDE];
      float a0 = 0.f, a1 = 0.f;
#pragma unroll
      for (int j = 0; j < Dn; j += 2) {
        float2 rv0 = *(const float2*)(r0 + j);
        float2 rv1 = *(const float2*)(r1 + j);
        float2 hv = *(const float2*)(hbuf + j);
        a0 += rv0.x * hv.x + rv0.y * hv.y;
        a1 += rv1.x * hv.x + rv1.y * hv.y;
      }
      // score_m = <Rh, item>  (lane-paired dot + wave reduction)
      float s = wred_sum(a0 * ia + a1 * ib);
      if (lane == m) myscore = s;
      // make sure our LDS reads retired before DMA overwrites this buffer
      asm volatile("s_wait_dscnt 0x0" ::: "memory");
      if (g + 3 < HOPn * Mn)
        tdm_load_mat64(rel + (size_t)mr[base0 + g + 3] * (Dn * Dn),
                       (unsigned)(uintptr_t)&rmat[(g + 3) % 3][0]);
    }
    // softmax over M=32 entirely in registers (lane m holds score[m])
    float mx = wred_max(myscore);
    float e = expf(myscore - mx);
    float myprob = e / wred_sum(e);
    // o = sum_m probs[m] * t[m][:]
    for (int m = 0; m < Mn; ++m) {
      float pm = __shfl(myprob, m, 32);
      const float* tp = ent + (size_t)mt[base0 + hop * Mn + m] * Dn;
      oa += pm * tp[lane];
      ob += pm * tp[lane + 32];
    }
    osa += oa; osb += ob;
    tmp[lane] = oa + ia;
    tmp[lane + 32] = ob + ib;
    __syncthreads();
    // item = lrelu((o + item) @ transform_w.T): lane owns rows {lane,lane+32}
    float na = 0.f, nb = 0.f;
    const float* twa = tw + (size_t)lane * Dn;
    const float* twb = tw + (size_t)(lane + 32) * Dn;
    for (int j = 0; j < Dn; ++j) {
      float t = tmp[j];
      na += twa[j] * t;
      nb += twb[j] * t;
    }
    ia = lrelu(na); ib = lrelu(nb);
    __syncthreads();
  }
  us[(size_t)bl * Dn + lane] = i0a + osa;
  us[(size_t)bl * Dn + lane + 32] = i0b + osb;
}

// ---------------- Kernel 2: sequential interest scan (WMMA GEMMs) ------
__global__ void k_scan(const float* __restrict__ uw, const int* __restrict__ user,
                       const float* __restrict__ w1g, const float* __restrict__ w2g,
                       const float* __restrict__ pe, const float* __restrict__ us,
                       float* __restrict__ xg) {
  __shared__ float w1[64 * 65], w2[64 * 65];
  __shared__ float sumb[16 * 65], prodb[16 * 65];
  __shared__ float t1b[8 * 65], t2b[8 * 65], carry[8 * 65];
  const int tid = threadIdx.x, wave = tid >> 5;   // 256 thr = 8 waves
  for (int idx = tid; idx < 4096; idx += 256) {
    int r = idx >> 6, c = idx & 63;
    w1[r * 65 + c] = w1g[idx];
    w2[r * 65 + c] = w2g[idx];
  }
  for (int idx = tid; idx < 16 * 64; idx += 256) {
    int r = idx >> 6, c = idx & 63;
    sumb[r * 65 + c] = 0.f; prodb[r * 65 + c] = 0.f;   // rows 8..15 stay zero
    if (r < Bn) carry[r * 65 + c] = uw[(size_t)user[r] * Dn + c];
  }
  __syncthreads();
  const int which = wave >> 2, ct = wave & 3;  // 2 GEMMs x 4 col-tiles
  for (int t = 0; t < Ln; ++t) {
    for (int idx = tid; idx < Bn * Dn; idx += 256) {
      int b = idx >> 6, d = idx & 63;
      float s = us[((size_t)b * Ln + t) * Dn + d];
      float c = carry[b * 65 + d];
      sumb[b * 65 + d] = s + c;
      prodb[b * 65 + d] = s * c;
    }
    __syncthreads();
    {
      const float* X = which ? prodb : sumb;
      const float* W = which ? w2 : w1;
      v8f acc = {};
      acc = wmma_xwT(X, 65, 0, W, 65, ct * 16, 64, acc);
      wmma_store(which ? t2b : t1b, 65, 0, ct * 16, acc, 8);
    }
    __syncthreads();
    for (int idx = tid; idx < Bn * Dn; idx += 256) {
      int b = idx >> 6, d = idx & 63;
      float o = lrelu(t1b[b * 65 + d]) + lrelu(t2b[b * 65 + d]);
      carry[b * 65 + d] = o;
      xg[((size_t)b * Ln + t) * Dn + d] = o * 8.0f + pe[t * Dn + d]; // *sqrt(D)+pos
    }
    __syncthreads();
  }
}

// layernorm one row (wave-per-row, lane owns cols {lane, lane+32})
__device__ __forceinline__ void ln_row(float* buf, int r, int lane,
                                       const float* sc, const float* bi) {
  float v0 = buf[r * 65 + lane], v1 = buf[r * 65 + lane + 32];
  float mean = wred_sum(v0 + v1) * (1.f / 64.f);
  float d0 = v0 - mean, d1 = v1 - mean;
  float var = wred_sum(d0 * d0 + d1 * d1) * (1.f / 64.f);
  float inv = 1.f / sqrtf(var + 1e-8f);
  buf[r * 65 + lane]      = d0 * inv * sc[lane]      + bi[lane];
  buf[r * 65 + lane + 32] = d1 * inv * sc[lane + 32] + bi[lane + 32];
}

// ---------------- Kernel 3: LN + QKV projections ------------------------
__global__ void k_qkv(const float* __restrict__ xg, const float* __restrict__ lns,
                      const float* __restrict__ lnb, const float* __restrict__ ipw,
                      const float* __restrict__ ipb, int blk,
                      float* __restrict__ Qg, float* __restrict__ qg,
                      float* __restrict__ kg, float* __restrict__ vg) {
  const int b = blockIdx.x, tid = threadIdx.x, lane = tid & 31, wave = tid >> 5;
  __shared__ float xs[64 * 65], Qs[64 * 65];
  for (int idx = tid; idx < 4096; idx += 256) {
    int r = idx >> 6, c = idx & 63;
    xs[r * 65 + c] = (r < Ln) ? xg[((size_t)b * Ln + r) * Dn + c] : 0.f;
    Qs[r * 65 + c] = 0.f;
  }
  __syncthreads();
  for (int r = wave; r < Ln; r += 8) {  // Q = LN(x)
    float v0 = xs[r * 65 + lane], v1 = xs[r * 65 + lane + 32];
    float mean = wred_sum(v0 + v1) * (1.f / 64.f);
    float d0 = v0 - mean, d1 = v1 - mean;
    float var = wred_sum(d0 * d0 + d1 * d1) * (1.f / 64.f);
    float inv = 1.f / sqrtf(var + 1e-8f);
    Qs[r * 65 + lane]      = d0 * inv * lns[blk * 64 + lane]      + lnb[blk * 64 + lane];
    Qs[r * 65 + lane + 32] = d1 * inv * lns[blk * 64 + lane + 32] + lnb[blk * 64 + lane + 32];
  }
  __syncthreads();
  const float* W = ipw + (size_t)blk * 3 * 64 * 64;
  const float* Bv = ipb + blk * 192;
  for (int t = wave; t < 48; t += 8) {  // 3 matrices x 16 tiles
    int mat = t >> 4, sub = t & 15, rt = sub >> 2, ctile = sub & 3;
    const float* X = (mat == 0) ? Qs : xs;  // q from LN(x); k,v from raw x
    v8f acc = {};
    acc = wmma_xwT(X, 65, rt * 16, W + (size_t)mat * 4096, 64, ctile * 16, 64, acc);
    int hi = lane >> 4, n = lane & 15;
    float bias = Bv[mat * 64 + ctile * 16 + n];
    float* Og = (mat == 0) ? qg : (mat == 1 ? kg : vg);
#pragma unroll
    for (int v = 0; v < 8; ++v) {
      int r = rt * 16 + v + 8 * hi;
      if (r < Ln) Og[((size_t)b * Ln + r) * Dn + ctile * 16 + n] = acc[v] + bias;
    }
  }
  __syncthreads();
  for (int idx = tid; idx < Ln * Dn; idx += 256) {
    int r = idx >> 6, c = idx & 63;
    Qg[((size_t)b * Ln + r) * Dn + c] = Qs[r * 65 + c];
  }
}

// ---------------- Kernel 4: causal attention per (b,h) ------------------
__global__ void k_attn(const float* __restrict__ qg, const float* __restrict__ kg,
                       const float* __restrict__ vg, float* __restrict__ cg) {
  const int b = blockIdx.x >> 1, h = blockIdx.x & 1;
  const int tid = threadIdx.x, lane = tid & 31, wave = tid >> 5; // 128 thr
  __shared__ float qh[64 * 33], kh[64 * 33], vh[64 * 33], sc[64 * 65];
  for (int idx = tid; idx < 64 * 32; idx += 128) {
    int r = idx >> 5, c = idx & 31;
    size_t g = ((size_t)b * Ln + r) * Dn + h * 32 + c;
    qh[r * 33 + c] = (r < Ln) ? qg[g] : 0.f;
    kh[r * 33 + c] = (r < Ln) ? kg[g] : 0.f;
    vh[r * 33 + c] = (r < Ln) ? vg[g] : 0.f;
  }
  __syncthreads();
  const float rscale = 0.17677669529663687f; // 1/sqrt(32)
  for (int t = wave; t < 16; t += 4) {       // scores = q @ k^T
    int rt = t >> 2, ctile = t & 3;
    v8f acc = {};
    acc = wmma_xwT(qh, 33, rt * 16, kh, 33, ctile * 16, 32, acc);
    int hi = lane >> 4, n = lane & 15;
#pragma unroll
    for (int v = 0; v < 8; ++v) {
      int r = rt * 16 + v + 8 * hi, cc = ctile * 16 + n;
      sc[r * 65 + cc] = (cc <= r && r < Ln) ? acc[v] * rscale : -1e30f;
    }
  }
  __syncthreads();
  for (int r = wave; r < Ln; r += 4) {       // row softmax
    float v0 = sc[r * 65 + lane], v1 = sc[r * 65 + lane + 32];
    float mx = wred_max(fmaxf(v0, v1));
    float e0 = expf(v0 - mx), e1 = expf(v1 - mx);
    float s = wred_sum(e0 + e1);
    sc[r * 65 + lane] = e0 / s; sc[r * 65 + lane + 32] = e1 / s;
  }
  for (int idx = tid; idx < 14 * 64; idx += 128) { // zero pad rows 50..63
    int r = 50 + (idx >> 6), c = idx & 63;
    sc[r * 65 + c] = 0.f;
  }
  __syncthreads();
  for (int t = wave; t < 8; t += 4) {        // ctx = attn @ v
    int rt = t >> 1, ctile = t & 1;
    v8f acc = {};
    acc = wmma_xw(sc, 65, rt * 16, vh, 33, ctile * 16, 64, acc);
    int hi = lane >> 4, n = lane & 15;
#pragma unroll
    for (int v = 0; v < 8; ++v) {
      int r = rt * 16 + v + 8 * hi;
      if (r < Ln) cg[((size_t)b * Ln + r) * Dn + h * 32 + ctile * 16 + n] = acc[v];
    }
  }
}

// ---------------- Kernel 5: out-proj + residual + LN + FFN --------------
__global__ void k_mha_ffn(const float* __restrict__ Qg, const float* __restrict__ cg,
                          const float* __restrict__ opw, const float* __restrict__ opb,
                          const float* __restrict__ lfs, const float* __restrict__ lfb,
                          const float* __restrict__ c1w, const float* __restrict__ c1b,
                          const float* __restrict__ c2w, const float* __restrict__ c2b,
                          int blk, float* __restrict__ xg) {
  const int b = blockIdx.x, tid = threadIdx.x, lane = tid & 31, wave = tid >> 5;
  __shared__ float cs[64 * 65], xs[64 * 65], hs[64 * 65];
  for (int idx = tid; idx < 4096; idx += 256) {
    int r = idx >> 6, c = idx & 63;
    cs[r * 65 + c] = (r < Ln) ? cg[((size_t)b * Ln + r) * Dn + c] : 0.f;
    xs[r * 65 + c] = (r < Ln) ? Qg[((size_t)b * Ln + r) * Dn + c] : 0.f;
    hs[r * 65 + c] = 0.f;
  }
  __syncthreads();
  const float* Wo = opw + (size_t)blk * 4096;
  for (int t = wave; t < 16; t += 8) {       // x = Q + (ctx @ Wo^T + bo)
    int rt = t >> 2, ctile = t & 3;
    v8f acc = {};
    acc = wmma_xwT(cs, 65, rt * 16, Wo, 64, ctile * 16, 64, acc);
    int hi = lane >> 4, n = lane & 15;
    float bias = opb[blk * 64 + ctile * 16 + n];
#pragma unroll
    for (int v = 0; v < 8; ++v) {
      int r = rt * 16 + v + 8 * hi;
      if (r < Ln) xs[r * 65 + ctile * 16 + n] += acc[v] + bias;
    }
  }
  __syncthreads();
  for (int r = wave; r < Ln; r += 8)         // LN fwd (in place)
    ln_row(xs, r, lane, lfs + blk * 64, lfb + blk * 64);
  __syncthreads();
  const float* W1 = c1w + (size_t)blk * 4096;
  for (int t = wave; t < 16; t += 8) {       // hdn = relu(x @ W1^T + b1)
    int rt = t >> 2, ctile = t & 3;
    v8f acc = {};
    acc = wmma_xwT(xs, 65, rt * 16, W1, 64, ctile * 16, 64, acc);
    int hi = lane >> 4, n = lane & 15;
    float bias = c1b[blk * 64 + ctile * 16 + n];
#pragma unroll
    for (int v = 0; v < 8; ++v) {
      int r = rt * 16 + v + 8 * hi;
      if (r < Ln) hs[r * 65 + ctile * 16 + n] = fmaxf(acc[v] + bias, 0.f);
    }
  }
  __syncthreads();
  const float* W2 = c2w + (size_t)blk * 4096;
  for (int t = wave; t < 16; t += 8) {       // x_out = x + hdn @ W2^T + b2
    int rt = t >> 2, ctile = t & 3;
    v8f acc = {};
    acc = wmma_xwT(hs, 65, rt * 16, W2, 64, ctile * 16, 64, acc);
    int hi = lane >> 4, n = lane & 15;
    float bias = c2b[blk * 64 + ctile * 16 + n];
#pragma unroll
    for (int v = 0; v < 8; ++v) {
      int r = rt * 16 + v + 8 * hi;
      if (r < Ln)
        xg[((size_t)b * Ln + r) * Dn + ctile * 16 + n] =
            xs[r * 65 + ctile * 16 + n] + acc[v] + bias;
    }
  }
}

// ---------------- Kernel 6: final LN + pos/neg logits -------------------
__global__ void k_logits(const float* __restrict__ xg, const float* __restrict__ lls,
                         const float* __restrict__ llb, const float* __restrict__ ent,
                         const int* __restrict__ pos, const int* __restrict__ neg,
                         float* __restrict__ out) {
  const int idx = blockIdx.x * 4 + (threadIdx.x >> 5); // wave per (b,l)
  const int lane = threadIdx.x & 31;
  if (idx >= Bn * Ln) return;
  float v0 = xg[(size_t)idx * Dn + lane], v1 = xg[(size_t)idx * Dn + lane + 32];
  float mean = wred_sum(v0 + v1) * (1.f / 64.f);
  float d0 = v0 - mean, d1 = v1 - mean;
  float var = wred_sum(d0 * d0 + d1 * d1) * (1.f / 64.f);
  float inv = 1.f / sqrtf(var + 1e-8f);
  float f0 = d0 * inv * lls[lane] + llb[lane];
  float f1 = d1 * inv * lls[lane + 32] + llb[lane + 32];
  const float* pe = ent + (size_t)pos[idx] * Dn;
  const float* ne = ent + (size_t)neg[idx] * Dn;
  float pd = wred_sum(f0 * pe[lane] + f1 * pe[lane + 32]);
  float nd = wred_sum(f0 * ne[lane] + f1 * ne[lane + 32]);
  if (lane == 0) { out[idx] = pd; out[Bn * Ln + idx] = nd; }
}

// ---------------- Host launcher -----------------------------------------
extern "C" void kernel_launch(void* const* d_in, const int* in_sizes, int n_in,
                              void* d_out, int out_size, void* d_ws, size_t ws_size,
                              hipStream_t stream) {
  (void)in_sizes; (void)n_in; (void)out_size; (void)ws_size;
  const float* user_embed_w    = (const float*)d_in[0];
  const float* entity_embed_w  = (const float*)d_in[1];
  const float* relation_embed_w= (const float*)d_in[2];
  const float* transform_w     = (const float*)d_in[3];
  const float* interest_w1     = (const float*)d_in[4];
  const float* interest_w2     = (const float*)d_in[5];
  const float* pos_embed_w     = (const float*)d_in[6];
  const float* ln_attn_scale   = (const float*)d_in[7];
  const float* ln_attn_bias    = (const float*)d_in[8];
  const float* in_proj_w       = (const float*)d_in[9];
  const float* in_proj_b       = (const float*)d_in[10];
  const float* out_proj_w      = (const float*)d_in[11];
  const float* out_proj_b      = (const float*)d_in[12];
  const float* ln_fwd_scale    = (const float*)d_in[13];
  const float* ln_fwd_bias     = (const float*)d_in[14];
  const float* conv1_w         = (const float*)d_in[15];
  const float* conv1_b         = (const float*)d_in[16];
  const float* conv2_w         = (const float*)d_in[17];
  const float* conv2_b         = (const float*)d_in[18];
  const float* last_ln_scale   = (const float*)d_in[19];
  const float* last_ln_bias    = (const float*)d_in[20];
  const int* user = (const int*)d_in[21];
  const int* seq  = (const int*)d_in[22];
  const int* pos  = (const int*)d_in[23];
  const int* neg  = (const int*)d_in[24];
  const int* mh   = (const int*)d_in[25];
  const int* mr   = (const int*)d_in[26];
  const int* mt   = (const int*)d_in[27];
  float* out = (float*)d_out;
  float* ws = (float*)d_ws;
  const size_t SEG = (size_t)Bn * Ln * Dn; // 25600 floats
  float* w_us = ws + 0 * SEG;
  float* w_x  = ws + 1 * SEG;
  float* w_Q  = ws + 2 * SEG;
  float* w_q  = ws + 3 * SEG;
  float* w_k  = ws + 4 * SEG;
  float* w_v  = ws + 5 * SEG;
  float* w_c  = ws + 6 * SEG;

  k_ripple<<<Bn * Ln, 32, 0, stream>>>(entity_embed_w, relation_embed_w,
                                       transform_w, seq, mh, mr, mt, w_us);
  k_scan<<<1, 256, 0, stream>>>(user_embed_w, user, interest_w1, interest_w2,
                                pos_embed_w, w_us, w_x);
  for (int blk = 0; blk < 2; ++blk) {
    k_qkv<<<Bn, 256, 0, stream>>>(w_x, ln_attn_scale, ln_attn_bias, in_proj_w,
                                  in_proj_b, blk, w_Q, w_q, w_k, w_v);
    k_attn<<<Bn * 2, 128, 0, stream>>>(w_q, w_k, w_v, w_c);
    k_mha_ffn<<<Bn, 256, 0, stream>>>(w_Q, w_c, out_proj_w, out_proj_b,
                                      ln_fwd_scale, ln_fwd_bias, conv1_w, conv1_b,
                                      conv2_w, conv2_b, blk, w_x);
  }
  k_logits<<<(Bn * Ln + 3) / 4, 128, 0, stream>>>(w_x, last_ln_scale, last_ln_bias,
                                                  entity_embed_w, pos, neg, out);
}